// DynamicKernel_2388001817009
// MI455X (gfx1250) — compile-verified
//
#include <hip/hip_runtime.h>

// ---------------------------------------------------------------------------
// Gated 5x5 conv (1->25 ch, tanh gate, windowed recombine) for 28x28 images.
// out[b,y,x] = sum_c P_T[c] * tanh(bias[c] + sum_k W[c*25+k] * P[k])
// where P[k=5u+v] = x[b, y+u-2, x+v-2] (zero padded), P_T[c=5i+j] = P[5j+i].
//
// GEMM per 16-pixel tile: A[16 x 28(K pad)] x B[28 x 32(N pad)] via
// V_WMMA_F32_16X16X4_F32 (7 K-steps x 2 N-tiles = 14 WMMA / tile).
// Image staged zero-padded (36x36) in LDS -> all gathers unconditional.
// Gate nonlinearity via gfx1250 hardware V_TANH_F32 when available.
// ---------------------------------------------------------------------------

typedef __attribute__((ext_vector_type(2))) float v2f;
typedef __attribute__((ext_vector_type(8))) float v8f;

#if __has_builtin(__builtin_amdgcn_tanhf)
#define TANH(v) __builtin_amdgcn_tanhf(v)
#else
#define TANH(v) tanhf(v)
#endif

#define KS    5
#define IMG   28
#define NPIX  784            // 28*28
#define PIMG  36             // 28 + 2*4 padding (matches reference xp)
#define PPIX  1296           // 36*36
#define NTILE 49             // 784 / 16, exact
#define NCH   25
#define WAVES 8

__global__ __launch_bounds__(256) void gated_conv_wmma(
    const float* __restrict__ x, const float* __restrict__ W,
    const float* __restrict__ bias, float* __restrict__ out) {
  __shared__ float sPad[PPIX];                 // zero-padded 36x36 image
  __shared__ float sW[NCH * NCH];              // 25x25 weights (row = channel)
  __shared__ float sB[32];                     // bias, padded with zeros
  __shared__ float sGate[WAVES][16][32];       // per-wave gate tile scratch

  const int tid  = threadIdx.x;
  const int wave = tid >> 5;
  const int lane = tid & 31;
  const int b    = blockIdx.x;

  // ---- staging: zero pad frame (b128), weights, bias -----------------------
  {
    const float4 z4 = make_float4(0.f, 0.f, 0.f, 0.f);
    for (int i = tid; i < PPIX / 4; i += 256) ((float4*)sPad)[i] = z4;
  }
  for (int i = tid; i < NCH * NCH; i += 256)  sW[i]   = W[i];
  if (tid < 32)                               sB[tid] = (tid < NCH) ? bias[tid] : 0.0f;
  __syncthreads();

  // interior fill, 16B vectorized: 196 float4 chunks, 7 per image row.
  const float4* __restrict__ xb4 = (const float4*)(x + (size_t)b * NPIX);
  for (int c = tid; c < NPIX / 4; c += 256) {
    const float4 v  = xb4[c];
    const int   row = c / 7;
    const int   cg  = (c % 7) * 4;
    *(float4*)&sPad[(row + 4) * PIMG + 4 + cg] = v;   // 16B-aligned in LDS
  }
  __syncthreads();

  const int col = lane & 15;   // N column within tile / A row (pixel) index
  const int hf  = lane >> 4;   // lane half selects K / M sub-blocks

  // ---- B fragments in registers: B[k][n] = W[n*25 + k] ---------------------
  // 16x16x4 f32 layout: B vgpr r holds K-row (r + 2*hf), N = col.
  v2f   bfrag[7][2];
  float biasv[2];
#pragma unroll
  for (int nt = 0; nt < 2; ++nt) {
    const int ch = nt * 16 + col;
    biasv[nt] = (ch < NCH) ? sB[ch] : 0.0f;
#pragma unroll
    for (int kg = 0; kg < 7; ++kg) {
      const int k0 = 4 * kg + 2 * hf;
      v2f bv;
      bv.x = (ch < NCH && k0     < NCH) ? sW[ch * NCH + k0]     : 0.0f;
      bv.y = (ch < NCH && k0 + 1 < NCH) ? sW[ch * NCH + k0 + 1] : 0.0f;
      bfrag[kg][nt] = bv;
    }
  }

  // ---- hoisted gather offsets (depend only on lane half) -------------------
  // A fragment: k-pair (4*kg + 2*hf, +1); off(k) = (k/5)*36 + k%5.
  int aoff[7][2];
#pragma unroll
  for (int kg = 0; kg < 7; ++kg) {
    const int k0 = 4 * kg + 2 * hf;
    aoff[kg][0] = (k0 / KS) * PIMG + (k0 % KS);
    aoff[kg][1] = ((k0 + 1) / KS) * PIMG + ((k0 + 1) % KS);
  }
  // Epilogue transposed patch: channel c=5i+j -> offset j*36 + i.
  int eoff[16];
#pragma unroll
  for (int cc = 0; cc < 16; ++cc) {
    const int c = hf * 16 + cc;
    eoff[cc] = (c % KS) * PIMG + (c / KS);
  }

  // ---- main loop: wave-uniform tile walk (EXEC all-ones at every WMMA) -----
  for (int tile = wave; tile < NTILE; tile += WAVES) {
    const int pbase = tile * 16;
    const int p     = pbase + col;                       // lane's pixel
    const int base  = (p / IMG + 2) * PIMG + (p % IMG + 2);  // padded center-2

    v8f acc0 = {};
    v8f acc1 = {};
#pragma unroll
    for (int kg = 0; kg < 7; ++kg) {
      v2f a;
      a.x = sPad[base + aoff[kg][0]];
      a.y = sPad[base + aoff[kg][1]];
      acc0 = __builtin_amdgcn_wmma_f32_16x16x4_f32(false, a, false, bfrag[kg][0],
                                                   (short)0, acc0, false, false);
      acc1 = __builtin_amdgcn_wmma_f32_16x16x4_f32(false, a, false, bfrag[kg][1],
                                                   (short)0, acc1, false, false);
    }

    // gate = tanh(acc + bias); C vgpr r holds M = r + 8*hf, N = col.
    // Channels >= 25 come out exactly tanh(0+0) = 0 (B columns zeroed).
#pragma unroll
    for (int r = 0; r < 8; ++r) {
      const int m = r + 8 * hf;
      sGate[wave][m][col]      = TANH(acc0[r] + biasv[0]);
      sGate[wave][m][16 + col] = TANH(acc1[r] + biasv[1]);
    }
    // Wave-internal LDS visibility: DS ops in-order per wave; wait completion.
    asm volatile("s_wait_dscnt 0" ::: "memory");

    // Epilogue: lane handles pixel m = col, channel half hf (16 channels,
    // zero-gate channels contribute nothing); combine halves via shfl.
    float sum = 0.0f;
#pragma unroll
    for (int cc = 0; cc < 16; ++cc)
      sum += sPad[base + eoff[cc]] * sGate[wave][col][hf * 16 + cc];
    sum += __shfl_xor(sum, 16, 32);
    if (hf == 0) out[(size_t)b * NPIX + p] = sum;
  }
}

extern "C" void kernel_launch(void* const* d_in, const int* in_sizes, int n_in,
                              void* d_out, int out_size, void* d_ws, size_t ws_size,
                              hipStream_t stream) {
  const float* x  = (const float*)d_in[0];  // [B,1,28,28]
  const float* W  = (const float*)d_in[1];  // [25,1,5,5]
  const float* bs = (const float*)d_in[2];  // [25]
  float* out = (float*)d_out;               // [B,1,28,28]
  const int B = in_sizes[0] / NPIX;         // 2048
  gated_conv_wmma<<<dim3(B), dim3(256), 0, stream>>>(x, W, bs, out);
}